// NodeModel_5428838662513
// MI455X (gfx1250) — compile-verified
//
#include <hip/hip_runtime.h>
#include <hip/hip_bf16.h>

#define N_NODES 100000
#define N_EDGES 1600000
#define NODE_IN 64
#define EDGE_OUT 48
#define HIDDEN 256
#define NODE_OUT 64
#define KDIM (NODE_IN + EDGE_OUT)   // 112

typedef __attribute__((ext_vector_type(2))) float v2f;
typedef __attribute__((ext_vector_type(8))) float v8f;

// ---------------- Kernel 1: zero workspace (agg sums + counts) ----------------
__global__ void init_ws_kernel(float* __restrict__ agg, float* __restrict__ cnt) {
    int stride = gridDim.x * blockDim.x;
    int i = blockIdx.x * blockDim.x + threadIdx.x;
    const int total = N_NODES * EDGE_OUT;
    for (int t = i; t < total; t += stride) agg[t] = 0.0f;
    for (int t = i; t < N_NODES; t += stride) cnt[t] = 0.0f;
}

// ---------------- Kernel 2: scatter-add edge features + counts ----------------
__global__ void scatter_edges_kernel(const float* __restrict__ edge_attr,
                                     const int* __restrict__ edge_src,
                                     float* __restrict__ agg,
                                     float* __restrict__ cnt) {
    const int items = N_EDGES * (EDGE_OUT / 4);       // 19.2M
    int stride = gridDim.x * blockDim.x;
    for (int i = blockIdx.x * blockDim.x + threadIdx.x; i < items; i += stride) {
        int e = i / (EDGE_OUT / 4);
        int g = i - e * (EDGE_OUT / 4);
        int node = edge_src[e];                        // row 0 of edge_index
        float4 v = ((const float4*)edge_attr)[i];
        float* dst = agg + node * EDGE_OUT + g * 4;
        atomicAdd(dst + 0, v.x);
        atomicAdd(dst + 1, v.y);
        atomicAdd(dst + 2, v.z);
        atomicAdd(dst + 3, v.w);
        if (g == 0) atomicAdd(cnt + node, 1.0f);
    }
}

// ---------------- Kernel 3: fused concat + MLP via V_WMMA_F32_16X16X4_F32 ----
#define WAVES_PER_BLOCK 8
#define ROWS_PER_WAVE 16
#define SA_STRIDE 114            // padded 112 (even for b64, bank-friendly)
#define ST_STRIDE 18             // padded 16 for the h-transpose staging
#define WAVE_LDS (16 * SA_STRIDE)          // 1824 floats per wave slice
#define W1K_STRIDE 114           // transposed W1 chunk: [128 cols][114]
#define W2K_STRIDE 130           // transposed W2 chunk: [64 cols][130]
#define SW_FLOATS (128 * W1K_STRIDE)       // 14592 floats (>= 64*130)

__global__ __launch_bounds__(256) void mlp_wmma_kernel(
    const float* __restrict__ x, const float* __restrict__ agg,
    const float* __restrict__ cnt,
    const float* __restrict__ W1, const float* __restrict__ b1,
    const float* __restrict__ W2, const float* __restrict__ b2,
    float* __restrict__ out) {
    // 14592 (shared weight chunk) + 8*1824 (per-wave slices) = 29184 floats = 114 KB
    __shared__ float smem[SW_FLOATS + WAVES_PER_BLOCK * WAVE_LDS];
    float* sW = smem;                                   // block-shared weight chunk
    const int lane = threadIdx.x & 31;
    const int wave = threadIdx.x >> 5;
    float* sA = smem + SW_FLOATS + wave * WAVE_LDS;     // private per-wave slice

    const int row0 = (blockIdx.x * WAVES_PER_BLOCK + wave) * ROWS_PER_WAVE;

    // ---- Stage A = [x | agg/max(cnt,1)] (16 x 112) into LDS (zero padded) ----
    for (int it = lane; it < 16 * (NODE_IN / 4); it += 32) {
        int r = it >> 4, c4 = it & 15;
        int row = row0 + r;
        float4 v = make_float4(0.f, 0.f, 0.f, 0.f);
        if (row < N_NODES) v = ((const float4*)x)[row * (NODE_IN / 4) + c4];
        float* p = sA + r * SA_STRIDE + c4 * 4;
        p[0] = v.x; p[1] = v.y; p[2] = v.z; p[3] = v.w;
    }
    for (int it = lane; it < 16 * (EDGE_OUT / 4); it += 32) {
        int r = it / (EDGE_OUT / 4);
        int c4 = it - r * (EDGE_OUT / 4);
        int row = row0 + r;
        float4 v = make_float4(0.f, 0.f, 0.f, 0.f);
        float inv = 1.0f;
        if (row < N_NODES) {
            v = ((const float4*)agg)[row * (EDGE_OUT / 4) + c4];
            inv = 1.0f / fmaxf(cnt[row], 1.0f);
        }
        float* p = sA + r * SA_STRIDE + NODE_IN + c4 * 4;
        p[0] = v.x * inv; p[1] = v.y * inv; p[2] = v.z * inv; p[3] = v.w * inv;
    }
    __asm__ volatile("s_wait_dscnt 0" ::: "memory");    // cross-lane LDS RAW in wave

    const int m  = lane & 15;       // A row (= C/D column)
    const int hw = lane >> 4;       // half-wave selector

    // ---- GEMM1: [16x112] @ [112x256] -> 16 accumulator tiles, 2 N-chunks ----
    v8f acc1[16] = {};
    for (int c = 0; c < 2; ++c) {
        __syncthreads();            // previous chunk fully consumed
        // cooperative transposed load of W1 columns [c*128, c*128+128)
        for (int it = threadIdx.x; it < KDIM * 128; it += 256) {
            int k = it >> 7;        // 0..111
            int n = it & 127;
            sW[n * W1K_STRIDE + k] = W1[k * HIDDEN + c * 128 + n];
        }
        __syncthreads();
        for (int j = 0; j < KDIM / 4; ++j) {
            const int kb = 4 * j + hw * 2;
            v2f a = *(const v2f*)(sA + m * SA_STRIDE + kb);
            #pragma unroll
            for (int tl = 0; tl < 8; ++tl) {
                v2f b = *(const v2f*)(sW + (tl * 16 + m) * W1K_STRIDE + kb);
                acc1[c * 8 + tl] = __builtin_amdgcn_wmma_f32_16x16x4_f32(
                    false, a, false, b, (short)0, acc1[c * 8 + tl], false, false);
            }
        }
    }

    // ---- bias + ReLU, transpose h through private LDS, GEMM2 in 2 K-chunks ----
    v8f acc2[4] = {};
    float* sT = sA;                 // reuse A slice (A no longer needed)
    for (int c = 0; c < 2; ++c) {
        __syncthreads();            // everyone done with previous sW contents
        // cooperative transposed load of W2 rows [c*128, c*128+128)
        for (int it = threadIdx.x; it < 128 * NODE_OUT; it += 256) {
            int k = it >> 6;        // 0..127
            int n = it & 63;
            sW[n * W2K_STRIDE + k] = W2[(c * 128 + k) * NODE_OUT + n];
        }
        __syncthreads();
        for (int tt = 0; tt < 8; ++tt) {
            const int t = c * 8 + tt;                   // global h col-tile
            const float bias = b1[t * 16 + m];
            #pragma unroll
            for (int r = 0; r < 8; ++r) {
                float h = fmaxf(acc1[t][r] + bias, 0.0f);
                sT[(hw * 8 + r) * ST_STRIDE + m] = h;   // C-layout -> row-major
            }
            __asm__ volatile("s_wait_dscnt 0" ::: "memory");
            #pragma unroll
            for (int j2 = 0; j2 < 4; ++j2) {
                const int kb  = 4 * j2 + hw * 2;        // K within 16-tile
                const int kbl = tt * 16 + kb;           // K within 128-chunk
                v2f a = *(const v2f*)(sT + m * ST_STRIDE + kb);
                #pragma unroll
                for (int n = 0; n < 4; ++n) {
                    v2f b = *(const v2f*)(sW + (n * 16 + m) * W2K_STRIDE + kbl);
                    acc2[n] = __builtin_amdgcn_wmma_f32_16x16x4_f32(
                        false, a, false, b, (short)0, acc2[n], false, false);
                }
            }
            // DS ops are in-order per wave: next tile's stores cannot pass loads.
        }
    }

    // ---- write out 16x64 tile (+b2), rows guarded ----
    #pragma unroll
    for (int n = 0; n < 4; ++n) {
        const float bias2 = b2[n * 16 + m];
        #pragma unroll
        for (int r = 0; r < 8; ++r) {
            int row = row0 + hw * 8 + r;
            if (row < N_NODES) out[row * NODE_OUT + n * 16 + m] = acc2[n][r] + bias2;
        }
    }
}

// ---------------- host-side launcher ----------------
extern "C" void kernel_launch(void* const* d_in, const int* in_sizes, int n_in,
                              void* d_out, int out_size, void* d_ws, size_t ws_size,
                              hipStream_t stream) {
    // setup_inputs order: x, edge_index, edge_attr, u, batch, W1, b1, W2, b2
    const float* x         = (const float*)d_in[0];
    const int*   edge_src  = (const int*)d_in[1];     // row 0 = first N_EDGES entries
    const float* edge_attr = (const float*)d_in[2];
    const float* W1        = (const float*)d_in[5];
    const float* b1        = (const float*)d_in[6];
    const float* W2        = (const float*)d_in[7];
    const float* b2        = (const float*)d_in[8];
    float* out = (float*)d_out;

    float* agg = (float*)d_ws;                         // N_NODES*EDGE_OUT floats
    float* cnt = agg + (size_t)N_NODES * EDGE_OUT;     // N_NODES floats

    init_ws_kernel<<<2048, 256, 0, stream>>>(agg, cnt);

    const int items = N_EDGES * (EDGE_OUT / 4);
    scatter_edges_kernel<<<(items + 255) / 256, 256, 0, stream>>>(
        edge_attr, edge_src, agg, cnt);

    const int rows_per_block = WAVES_PER_BLOCK * ROWS_PER_WAVE;   // 128
    mlp_wmma_kernel<<<(N_NODES + rows_per_block - 1) / rows_per_block, 256, 0, stream>>>(
        x, agg, cnt, W1, b1, W2, b2, out);
}